// AI4DEM_19146964205924
// MI455X (gfx1250) — compile-verified
//
#include <hip/hip_runtime.h>
#include <cmath>

// ---------------- problem constants (match reference) ----------------
#define NG   2000
#define NNE  (NG * NG)          // 4,000,000 cells
#define KN_  500000.0f
#define DT_  1e-4f
#define EPS_ 1e-4f
#define GRAV 9.8f

// ---------------- tiling ----------------
#define TS   16                 // output tile is 16x16 cells (1 cell/thread, 256 thr)
#define HALO 2                  // 5x5 stencil
#define LTS  (TS + 2 * HALO)    // 20
#define LST  24                 // padded LDS row stride (dwords)

// ---- CDNA5 async global->LDS helpers (builtin confirmed present; it wants
// ---- non-const int* params per the round-1 diagnostic) ----
typedef int __attribute__((address_space(1))) gint_as1;
typedef int __attribute__((address_space(3))) lint_as3;

__device__ __forceinline__ void async_ld_b32(const float* gsrc, float* ldst) {
#if defined(__gfx1250__) && __has_builtin(__builtin_amdgcn_global_load_async_to_lds_b32)
    gint_as1* gp = (gint_as1*)(unsigned long long)gsrc;
    lint_as3* lp = (lint_as3*)(unsigned int)(unsigned long long)ldst;
    __builtin_amdgcn_global_load_async_to_lds_b32(gp, lp, 0, 0);
#else
    *ldst = *gsrc;              // fallback: plain load + ds_store
#endif
}

__device__ __forceinline__ void wait_async_all() {
#if defined(__gfx1250__) && __has_builtin(__builtin_amdgcn_s_wait_asynccnt)
    __builtin_amdgcn_s_wait_asynccnt(0);
#elif defined(__AMDGCN__)
    asm volatile("s_wait_asynccnt 0" ::: "memory");
#endif
}

__device__ __forceinline__ int clampi(int v, int lo, int hi) {
    return v < lo ? lo : (v > hi ? hi : v);
}

// ============ kernel 1: forces + integration (LDS-tiled stencil) ============
__global__ __launch_bounds__(256)
void dem_force(const float* __restrict__ xg,  const float* __restrict__ yg,
               const float* __restrict__ vxg, const float* __restrict__ vyg,
               const float* __restrict__ mg,
               float* __restrict__ xn,  float* __restrict__ yn,
               float* __restrict__ vxn, float* __restrict__ vyn,
               float* __restrict__ ox,  float* __restrict__ oy,
               float* __restrict__ ovx, float* __restrict__ ovy,
               float* __restrict__ om,
               float eta)
{
    __shared__ float sx [LTS * LST];
    __shared__ float sy [LTS * LST];
    __shared__ float svx[LTS * LST];
    __shared__ float svy[LTS * LST];

    const int tid = threadIdx.x;
    const int gx0 = blockIdx.x * TS - HALO;
    const int gy0 = blockIdx.y * TS - HALO;

    // Stage tile + halo into LDS via the async data mover (ASYNCcnt-tracked).
    // Halo clamp to [0,N-1] matches roll semantics: edge rows/cols are all-zero
    // (particles live at indices >= 2), so clamp == wrap for the result.
    for (int idx = tid; idx < LTS * LTS; idx += 256) {
        const int lr = idx / LTS;
        const int lc = idx - lr * LTS;
        const int gr = clampi(gy0 + lr, 0, NG - 1);
        const int gc = clampi(gx0 + lc, 0, NG - 1);
        const int go = gr * NG + gc;
        const int lo = lr * LST + lc;
        async_ld_b32(xg  + go, sx  + lo);
        async_ld_b32(yg  + go, sy  + lo);
        async_ld_b32(vxg + go, svx + lo);
        async_ld_b32(vyg + go, svy + lo);
    }
    wait_async_all();
    __syncthreads();

    const int tx = tid & (TS - 1);
    const int ty = tid >> 4;
    const int lr = ty + HALO, lc = tx + HALO;
    const int gx = blockIdx.x * TS + tx;
    const int gy = blockIdx.y * TS + ty;
    const int g  = gy * NG + gx;          // NG % TS == 0 -> always in range

    const float x  = sx [lr * LST + lc];
    const float y  = sy [lr * LST + lc];
    const float vx = svx[lr * LST + lc];
    const float vy = svy[lr * LST + lc];

    float fxc = 0.f, fyc = 0.f, fxd = 0.f, fyd = 0.f;
#pragma unroll
    for (int di = -HALO; di <= HALO; ++di) {
#pragma unroll
        for (int dj = -HALO; dj <= HALO; ++dj) {
            const int   o    = (lr + di) * LST + (lc + dj);
            const float dxv  = x - sx[o];
            const float dyv  = y - sy[o];
            float d2 = fmaf(dxv, dxv, dyv * dyv);
            d2 = fmaxf(d2, EPS_ * EPS_);
            const float rinv = __frsqrt_rn(d2);   // == 1/dist (dist >= EPS)
            const float dist = d2 * rinv;
            const float sel  = (dist < 2.0f) ? 1.0f : 0.0f;
            const float fn   = KN_ * (dist - 2.0f) * rinv * sel;
            fxc = fmaf(fn, dxv, fxc);
            fyc = fmaf(fn, dyv, fyc);
            const float dvx = vx - svx[o];
            const float dvy = vy - svy[o];
            const float vn  = fmaf(dvx, dxv, dvy * dyv) * rinv;
            const float fd  = eta * vn * rinv * sel;
            fxd = fmaf(fd, dxv, fxd);
            fyd = fmaf(fd, dyv, fyd);
        }
    }

    // wall indicators + forces
    const float m   = mg[g];
    const float DSZ = (float)NG;
    const float il = (x != 0.0f && x < 1.0f) ? m : 0.0f;
    const float ir = (x > DSZ - 2.0f)        ? m : 0.0f;
    const float ib = (y != 0.0f && y < 1.0f) ? m : 0.0f;
    const float it = (y > DSZ - 2.0f)        ? m : 0.0f;
    const float fxl = KN_ * il * (1.0f - x);
    const float fxr = KN_ * ir * (x - DSZ + 2.0f);
    const float fyb = KN_ * ib * (1.0f - y);
    const float fyt = KN_ * it * (y - DSZ + 2.0f);

    const float vx1 = vx + DT_ * (fxl - fxr - fxc - fxd - eta * vx * (il + ir)) * m;
    const float vy1 = vy + DT_ * (-GRAV) * m
                         + DT_ * (fyb - fyt - fyc - fyd - eta * vy * (ib + it)) * m;
    const float x1  = fmaf(DT_, vx1, x);
    const float y1  = fmaf(DT_, vy1, y);

    // workspace (needed by the scatter pass)
    xn[g] = x1;  yn[g] = y1;  vxn[g] = vx1;  vyn[g] = vy1;
    // rebin phase 0: output starts as the pre-scatter updated state
    ox[g] = x1;  oy[g] = y1;  ovx[g] = vx1;  ovy[g] = vy1;  om[g] = m;
}

// ============ kernel 2: rebin phase 1 — zero all old cells first ============
__global__ __launch_bounds__(256)
void rebin_zero(const float* __restrict__ xg, const float* __restrict__ yg,
                float* __restrict__ ox,  float* __restrict__ oy,
                float* __restrict__ ovx, float* __restrict__ ovy,
                float* __restrict__ om)
{
    const int k = blockIdx.x * 256 + threadIdx.x;
    if (k >= NNE) return;
    const int cxo = clampi(__float2int_rn(xg[k]), 0, NG - 1);
    const int cyo = clampi(__float2int_rn(yg[k]), 0, NG - 1);
    if (cxo != 0 && cyo != 0) {
        const int oi = cyo * NG + cxo;
        ox[oi] = 0.f; oy[oi] = 0.f; ovx[oi] = 0.f; ovy[oi] = 0.f; om[oi] = 0.f;
    }
}

// ============ kernel 3: rebin phase 2 — set new cells from old-cell values ==
__global__ __launch_bounds__(256)
void rebin_set(const float* __restrict__ xg,  const float* __restrict__ yg,
               const float* __restrict__ xn,  const float* __restrict__ yn,
               const float* __restrict__ vxn, const float* __restrict__ vyn,
               float* __restrict__ ox,  float* __restrict__ oy,
               float* __restrict__ ovx, float* __restrict__ ovy,
               float* __restrict__ om)
{
    const int k = blockIdx.x * 256 + threadIdx.x;
    if (k >= NNE) return;
    const int cxo = clampi(__float2int_rn(xg[k]), 0, NG - 1);
    const int cyo = clampi(__float2int_rn(yg[k]), 0, NG - 1);
    if (cxo == 0 || cyo == 0) return;                 // valid_old
    const int cxn = clampi(__float2int_rn(xn[k]), 0, NG - 1);
    const int cyn = clampi(__float2int_rn(yn[k]), 0, NG - 1);
    if (cxn == 0 || cyn == 0) return;                 // valid_new
    const int oi = cyo * NG + cxo;                    // gat_idx (pre-zero gather)
    const int ni = cyn * NG + cxn;                    // new_idx
    ox[ni]  = xn[oi];
    oy[ni]  = yn[oi];
    ovx[ni] = vxn[oi];
    ovy[ni] = vyn[oi];
    om[ni]  = 1.0f;
}

// ================================ launch ================================
extern "C" void kernel_launch(void* const* d_in, const int* in_sizes, int n_in,
                              void* d_out, int out_size, void* d_ws, size_t ws_size,
                              hipStream_t stream)
{
    (void)in_sizes; (void)n_in; (void)out_size; (void)ws_size;

    const float* xg  = (const float*)d_in[0];
    const float* yg  = (const float*)d_in[1];
    const float* vxg = (const float*)d_in[2];
    const float* vyg = (const float*)d_in[3];
    const float* mg  = (const float*)d_in[4];

    float* out = (float*)d_out;
    const size_t NNs = (size_t)NNE;
    float* ox  = out;
    float* oy  = out + NNs;
    float* ovx = out + 2 * NNs;
    float* ovy = out + 3 * NNs;
    float* om  = out + 4 * NNs;

    float* ws  = (float*)d_ws;            // 4 * 16 MB scratch
    float* xn  = ws;
    float* yn  = ws + NNs;
    float* vxn = ws + 2 * NNs;
    float* vyn = ws + 3 * NNs;

    const double alpha = -std::log(0.7) / 3.14159265358979323846;
    const double gam   = alpha / std::sqrt(alpha * alpha + 1.0);
    const float  eta   = (float)(2.0 * gam * std::sqrt(500000.0 * 1.0));

    dim3 gridF(NG / TS, NG / TS);         // 125 x 125
    dem_force<<<gridF, dim3(256), 0, stream>>>(xg, yg, vxg, vyg, mg,
                                               xn, yn, vxn, vyn,
                                               ox, oy, ovx, ovy, om, eta);

    const int blocks1d = (NNE + 255) / 256;
    rebin_zero<<<blocks1d, 256, 0, stream>>>(xg, yg, ox, oy, ovx, ovy, om);
    rebin_set <<<blocks1d, 256, 0, stream>>>(xg, yg, xn, yn, vxn, vyn,
                                             ox, oy, ovx, ovy, om);
}